// hyperbolicGCN_hypergraph_77266461655829
// MI455X (gfx1250) — compile-verified
//
#include <hip/hip_runtime.h>

#define N_NODES 100000
#define N_EDGES 20000
#define NNZ     640000
#define FD      128

typedef __attribute__((ext_vector_type(2))) float v2f;
typedef __attribute__((ext_vector_type(8))) float v8f;

// ---------------------------------------------------------------------------
// Wave-wide (32-lane) sum, broadcast to all lanes.
// Stage 1 runs through the f32 matrix pipe: V_WMMA_F32_16X16X4_F32 with
// A = ones(16x4) and B carrying the 32 lane values in its two valid K-rows
// (B vgpr0: lanes 0-15 -> row K=0, lanes 16-31 -> row K=1), so
// D[m][n] = v_n + v_{n+16} for every m; every lane's d[0] = v_{lane&15} +
// v_{(lane&15)+16}. Four xor-shuffles finish the 16-value reduction.
// EXEC must be all ones at every call site (guaranteed: wave-per-row layout,
// grids divide exactly: 100000 = 12500 blocks * 8 waves, 640000 = 80000 * 8).
// ---------------------------------------------------------------------------
__device__ __forceinline__ float wave_sum(float v) {
  v2f a; a[0] = 1.0f; a[1] = 1.0f;
  v2f b; b[0] = v;    b[1] = 0.0f;
  v8f c = {};
  c = __builtin_amdgcn_wmma_f32_16x16x4_f32(false, a, false, b, (short)0, c,
                                            false, false);
  float p = c[0];
  p += __shfl_xor(p, 1);
  p += __shfl_xor(p, 2);
  p += __shfl_xor(p, 4);
  p += __shfl_xor(p, 8);
  return p;
}

__device__ __forceinline__ float sp_(float v) {          // softplus
  return v > 20.0f ? v : log1pf(expf(v));
}
__device__ __forceinline__ float arcosh_(float x) {
  return logf(x + sqrtf(fmaxf(x * x - 1.0f, 1e-7f)));
}

// ---------------------------------------------------------------------------
// h = proj(expmap0(proj_tan0([0, x]), c0), c0)      (x rows have 127 features)
// ---------------------------------------------------------------------------
__global__ __launch_bounds__(256) void k_init(const float* __restrict__ x,
                                              const float* __restrict__ curv,
                                              float* __restrict__ h) {
  int wid  = (blockIdx.x * blockDim.x + threadIdx.x) >> 5;
  int lane = threadIdx.x & 31;
  if (wid >= N_NODES) return;
  float c0 = sp_(curv[0]);
  float K = 1.0f / c0, sqrtK = sqrtf(K);
  const float* row = x + (size_t)wid * 127;
  int j = lane * 4;                       // components j .. j+3
  int i0 = (lane == 0) ? 0 : (j - 1);     // clamped: never reads row[-1]
  float u0 = row[i0];
  if (lane == 0) u0 = 0.0f;
  float u1 = row[j + 0], u2 = row[j + 1], u3 = row[j + 2];
  float s  = wave_sum(u0 * u0 + u1 * u1 + u2 * u2 + u3 * u3);
  float xn = fmaxf(sqrtf(s), 1e-5f);
  float sh = sinhf(fminf(xn / sqrtK, 15.0f));
  float sc = sqrtK * sh / xn;
  float r0 = sc * u0, r1 = sc * u1, r2 = sc * u2, r3 = sc * u3;   // r0(lane0)=0
  float ysq = wave_sum(r0 * r0 + r1 * r1 + r2 * r2 + r3 * r3);
  float x0 = sqrtf(fmaxf(K + ysq, 1e-7f));
  if (lane == 0) r0 = x0;
  ((float4*)(h + (size_t)wid * FD))[lane] = make_float4(r0, r1, r2, r3);
}

// ---------------------------------------------------------------------------
// dst = proj_tan0(logmap0(src, softplus(curv[ci])))   (also used for output)
// ---------------------------------------------------------------------------
__global__ __launch_bounds__(256) void k_logmap_tan0(const float* __restrict__ src,
                                                     float* __restrict__ dst,
                                                     const float* __restrict__ curv,
                                                     int ci) {
  int wid  = (blockIdx.x * blockDim.x + threadIdx.x) >> 5;
  int lane = threadIdx.x & 31;
  if (wid >= N_NODES) return;
  float c = sp_(curv[ci]);
  float K = 1.0f / c, sqrtK = sqrtf(K);
  float4 v = ((const float4*)(src + (size_t)wid * FD))[lane];
  float a0 = (lane == 0) ? 0.0f : v.x;
  float ysq = wave_sum(a0 * a0 + v.y * v.y + v.z * v.z + v.w * v.w);
  float yn = fmaxf(sqrtf(ysq), 1e-5f);
  float x0 = __shfl(v.x, 0);
  float th = fmaxf(x0 / sqrtK, 1.0f + 1e-7f);
  float sc = sqrtK * arcosh_(th) / yn;
  float o0 = (lane == 0) ? 0.0f : sc * v.x;
  ((float4*)(dst + (size_t)wid * FD))[lane] =
      make_float4(o0, sc * v.y, sc * v.z, sc * v.w);
}

// ---------------------------------------------------------------------------
// Degree computation (indices are layer-invariant -> done once per call)
// ---------------------------------------------------------------------------
__global__ __launch_bounds__(256) void k_count(const int* __restrict__ nidx,
                                               const int* __restrict__ eidx,
                                               float* __restrict__ Dc,
                                               float* __restrict__ Bc) {
  int i = blockIdx.x * blockDim.x + threadIdx.x;
  if (i < NNZ) {
    atomicAdd(&Dc[nidx[i]], 1.0f);
    atomicAdd(&Bc[eidx[i]], 1.0f);
  }
}
__global__ __launch_bounds__(256) void k_inv(float* __restrict__ p, int n) {
  int i = blockIdx.x * blockDim.x + threadIdx.x;
  if (i < n) { float v = p[i]; p[i] = v > 0.0f ? 1.0f / v : 0.0f; }
}

// ---------------------------------------------------------------------------
// Segment sums: wave-per-nnz, one 512B row per wave, f32 atomics into L2.
// wid is forced SGPR-uniform via readfirstlane so the index/degree loads
// become scalar (SMEM) loads instead of 32 redundant VMEM lane loads.
// Index streams are sequential per wave-group -> fixed-distance prefetch.
// ---------------------------------------------------------------------------
__global__ __launch_bounds__(256) void k_scatter_edge(const float* __restrict__ t,
                                                      const int* __restrict__ nidx,
                                                      const int* __restrict__ eidx,
                                                      const float* __restrict__ Binv,
                                                      float* __restrict__ efeat) {
  int wid  = (blockIdx.x * blockDim.x + threadIdx.x) >> 5;
  int lane = threadIdx.x & 31;
  if (wid >= NNZ) return;
  int wids = __builtin_amdgcn_readfirstlane(wid);
  if (wids + 4096 < NNZ) {
    __builtin_prefetch(&nidx[wids + 4096], 0, 0);
    __builtin_prefetch(&eidx[wids + 4096], 0, 0);
  }
  int n = nidx[wids], e = eidx[wids];
  float b = Binv[e];
  float4 v = ((const float4*)(t + (size_t)n * FD))[lane];
  float* dst = efeat + (size_t)e * FD + lane * 4;
  atomicAdd(dst + 0, b * v.x);
  atomicAdd(dst + 1, b * v.y);
  atomicAdd(dst + 2, b * v.z);
  atomicAdd(dst + 3, b * v.w);
}

__global__ __launch_bounds__(256) void k_scatter_node(const float* __restrict__ efeat,
                                                      const int* __restrict__ nidx,
                                                      const int* __restrict__ eidx,
                                                      const float* __restrict__ Dinv,
                                                      float* __restrict__ outb) {
  int wid  = (blockIdx.x * blockDim.x + threadIdx.x) >> 5;
  int lane = threadIdx.x & 31;
  if (wid >= NNZ) return;
  int wids = __builtin_amdgcn_readfirstlane(wid);
  if (wids + 4096 < NNZ) {
    __builtin_prefetch(&nidx[wids + 4096], 0, 0);
    __builtin_prefetch(&eidx[wids + 4096], 0, 0);
  }
  int n = nidx[wids], e = eidx[wids];
  float d = Dinv[n];
  float4 v = ((const float4*)(efeat + (size_t)e * FD))[lane];
  float* dst = outb + (size_t)n * FD + lane * 4;
  atomicAdd(dst + 0, d * v.x);
  atomicAdd(dst + 1, d * v.y);
  atomicAdd(dst + 2, d * v.z);
  atomicAdd(dst + 3, d * v.w);
}

// ---------------------------------------------------------------------------
// u_bias = logmap0(proj(expmap0(proj_tan0(bias), c_in), c_in), c_in)
// (mobius_add only consumes the bias through logmap0). Single full wave.
// ---------------------------------------------------------------------------
__global__ void k_bias(const float* __restrict__ biases,
                       const float* __restrict__ curv, int layer,
                       float* __restrict__ ubias) {
  int lane = threadIdx.x;                 // 32 threads, 1 wave, EXEC full
  const float* bias = biases + (size_t)layer * FD;
  float c = sp_(curv[layer]);
  float K = 1.0f / c, sqrtK = sqrtf(K);
  int j = lane * 4;
  float b0 = bias[j + 0];
  if (lane == 0) b0 = 0.0f;
  float b1 = bias[j + 1], b2 = bias[j + 2], b3 = bias[j + 3];
  float s  = wave_sum(b0 * b0 + b1 * b1 + b2 * b2 + b3 * b3);
  float xn = fmaxf(sqrtf(s), 1e-5f);
  float sh = sinhf(fminf(xn / sqrtK, 15.0f));
  float sc = sqrtK * sh / xn;
  float r0 = sc * b0, r1 = sc * b1, r2 = sc * b2, r3 = sc * b3;
  float ysq = wave_sum(r0 * r0 + r1 * r1 + r2 * r2 + r3 * r3);
  float x0 = sqrtf(fmaxf(K + ysq, 1e-7f));
  float yn2 = fmaxf(sqrtf(ysq), 1e-5f);
  float th2 = fmaxf(x0 / sqrtK, 1.0f + 1e-7f);
  float sc2 = sqrtK * arcosh_(th2) / yn2;
  float o0 = (lane == 0) ? 0.0f : sc2 * r0;
  ((float4*)ubias)[lane] = make_float4(o0, sc2 * r1, sc2 * r2, sc2 * r3);
}

// ---------------------------------------------------------------------------
// Fused tail of hyp_block:
// h = proj(expmap0(proj_tan0(leakyrelu(logmap0(proj(mobius_add(
//        proj(out), hyp_bias, cin)), cin))), cout), cout)
// mobius_add expanded with u_bias precomputed; everything in registers.
// ---------------------------------------------------------------------------
__global__ __launch_bounds__(256) void k_fused(const float* __restrict__ outb,
                                               const float* __restrict__ ubias,
                                               const float* __restrict__ curv,
                                               int layer, float* __restrict__ h) {
  int wid  = (blockIdx.x * blockDim.x + threadIdx.x) >> 5;
  int lane = threadIdx.x & 31;
  if (wid >= N_NODES) return;
  float cin = sp_(curv[layer]), cout = sp_(curv[layer + 1]);
  float K = 1.0f / cin,  sqrtK  = sqrtf(K);
  float Ko = 1.0f / cout, sqrtKo = sqrtf(Ko);
  float4 o = ((const float4*)(outb + (size_t)wid * FD))[lane];
  float4 u = ((const float4*)ubias)[lane];
  float om = (lane == 0) ? 0.0f : o.x;       // y components (comp0 masked)
  float um = (lane == 0) ? 0.0f : u.x;
  // res = proj(out, cin)
  float ysq = wave_sum(om * om + o.y * o.y + o.z * o.z + o.w * o.w);
  float x0 = sqrtf(fmaxf(K + ysq, 1e-7f));
  float yn = fmaxf(sqrtf(ysq), 1e-5f);
  // ptransp0(res, u, cin):  resv_j = u_j - alpha*(sqrtK - x0)*y_hat_j
  float dyu = wave_sum(om * um + o.y * u.y + o.z * u.z + o.w * u.w);
  float alpha = dyu / (yn * sqrtK);
  float f = alpha * (sqrtK - x0) / yn;
  float rvx = (lane == 0) ? 0.0f : (u.x - f * o.x);
  float rvy = u.y - f * o.y, rvz = u.z - f * o.z, rvw = u.w - f * o.w;
  // proj_tan(resv, res): w0 = <y, resv_y>/max(x0,eps); w_j = resv_j
  float ux = wave_sum(om * rvx + o.y * rvy + o.z * rvz + o.w * rvw);
  float w0 = ux / fmaxf(x0, 1e-7f);
  // expmap(w, res, cin): minkowski dot = sum(w^2) - 2*w0^2 = sum_{j>=1} - w0^2
  float swq = wave_sum(rvx * rvx + rvy * rvy + rvz * rvz + rvw * rvw);
  float normu = fminf(sqrtf(fmaxf(swq - w0 * w0, 1e-7f)), 1e6f);
  float theta = fmaxf(normu / sqrtK, 1e-5f);
  float thc = fminf(theta, 15.0f);
  float ch = coshf(thc), g = sinhf(thc) / theta;
  float tx = ch * om  + g * rvx;             // comp0 discarded by proj
  float ty = ch * o.y + g * rvy;
  float tz = ch * o.z + g * rvz;
  float tw = ch * o.w + g * rvw;
  float ysq2 = wave_sum(tx * tx + ty * ty + tz * tz + tw * tw);
  float x02 = sqrtf(fmaxf(K + ysq2, 1e-7f));
  // logmap0 + leaky_relu(0.01) (comp0 stays 0)
  float yn2 = fmaxf(sqrtf(ysq2), 1e-5f);
  float th2 = fmaxf(x02 / sqrtK, 1.0f + 1e-7f);
  float sc2 = sqrtK * arcosh_(th2) / yn2;
  float lx = sc2 * tx, ly = sc2 * ty, lz = sc2 * tz, lw = sc2 * tw;
  lx = lx >= 0.0f ? lx : 0.01f * lx;  ly = ly >= 0.0f ? ly : 0.01f * ly;
  lz = lz >= 0.0f ? lz : 0.01f * lz;  lw = lw >= 0.0f ? lw : 0.01f * lw;
  // expmap0(cout) + proj(cout)
  float nl2 = wave_sum(lx * lx + ly * ly + lz * lz + lw * lw);
  float nl = fmaxf(sqrtf(nl2), 1e-5f);
  float sh3 = sinhf(fminf(nl / sqrtKo, 15.0f));
  float sc3 = sqrtKo * sh3 / nl;
  float hx = sc3 * lx, hy = sc3 * ly, hz = sc3 * lz, hw = sc3 * lw;
  float ysq3 = wave_sum(hx * hx + hy * hy + hz * hz + hw * hw);
  float h0 = sqrtf(fmaxf(Ko + ysq3, 1e-7f));
  if (lane == 0) hx = h0;
  ((float4*)(h + (size_t)wid * FD))[lane] = make_float4(hx, hy, hz, hw);
}

// ---------------------------------------------------------------------------
extern "C" void kernel_launch(void* const* d_in, const int* in_sizes, int n_in,
                              void* d_out, int out_size, void* d_ws, size_t ws_size,
                              hipStream_t stream) {
  const float* x      = (const float*)d_in[0];
  const int*   nidx   = (const int*)d_in[1];          // hyperedge_index[0]
  const int*   eidx   = ((const int*)d_in[1]) + NNZ;  // hyperedge_index[1]
  const float* curv   = (const float*)d_in[2];
  const float* biases = (const float*)d_in[3];
  float* h  = (float*)d_out;                          // (N_NODES, 128) lives here

  // Workspace layout (~62 MB): t/out buffer reused, e_feat, degrees, u_bias.
  float* bufT  = (float*)d_ws;                        // 12,800,000 f
  float* bufE  = bufT + (size_t)N_NODES * FD;         //  2,560,000 f
  float* Dinv  = bufE + (size_t)N_EDGES * FD;         //    100,000 f
  float* Binv  = Dinv + N_NODES;                      //     20,000 f
  float* ubias = Binv + N_EDGES;                      //        128 f

  const int TPB = 256;
  const int nodeBlocks = (N_NODES + 7) / 8;           // 8 waves/block, exact
  const int nnzBlocks  = (NNZ + 7) / 8;               // exact

  // Degrees (indices identical for both layers).
  hipMemsetAsync(Dinv, 0, (size_t)(N_NODES + N_EDGES) * sizeof(float), stream);
  k_count<<<(NNZ + TPB - 1) / TPB, TPB, 0, stream>>>(nidx, eidx, Dinv, Binv);
  k_inv<<<(N_NODES + TPB - 1) / TPB, TPB, 0, stream>>>(Dinv, N_NODES);
  k_inv<<<(N_EDGES + TPB - 1) / TPB, TPB, 0, stream>>>(Binv, N_EDGES);

  // h = proj(expmap0(proj_tan0([0,x]), c0), c0)
  k_init<<<nodeBlocks, TPB, 0, stream>>>(x, curv, h);

  for (int layer = 0; layer < 2; ++layer) {
    k_bias<<<1, 32, 0, stream>>>(biases, curv, layer, ubias);
    // t = proj_tan0(logmap0(h, c_in))
    k_logmap_tan0<<<nodeBlocks, TPB, 0, stream>>>(h, bufT, curv, layer);
    // e_feat = segsum_edge(B[e] * t[n])
    hipMemsetAsync(bufE, 0, (size_t)N_EDGES * FD * sizeof(float), stream);
    k_scatter_edge<<<nnzBlocks, TPB, 0, stream>>>(bufT, nidx, eidx, Binv, bufE);
    // out = segsum_node(D[n] * e_feat[e])   (reuse bufT)
    hipMemsetAsync(bufT, 0, (size_t)N_NODES * FD * sizeof(float), stream);
    k_scatter_node<<<nnzBlocks, TPB, 0, stream>>>(bufE, nidx, eidx, Dinv, bufT);
    // h = fused bias/mobius/activation/curvature-switch
    k_fused<<<nodeBlocks, TPB, 0, stream>>>(bufT, ubias, curv, layer, h);
  }
  // final: proj_tan0(logmap0(h, c_last)) in place in d_out
  k_logmap_tan0<<<nodeBlocks, TPB, 0, stream>>>(h, h, curv, 2);
}